// ProsodyPredictorV15_10797547782207
// MI455X (gfx1250) — compile-verified
//
#include <hip/hip_runtime.h>
#include <math.h>

typedef float v2f __attribute__((ext_vector_type(2)));
typedef float v8f __attribute__((ext_vector_type(8)));
typedef unsigned int v4u __attribute__((ext_vector_type(4)));
typedef int v4i __attribute__((ext_vector_type(4)));
typedef int v8i __attribute__((ext_vector_type(8)));

#define B_N   4
#define T_LEN 2048
#define E_DIM 512
#define D_DIM 512
#define N_ST  16
#define L_NUM 4
#define M_ROWS (B_N * T_LEN)   // 8192

// ---------------------------------------------------------------------------
// Kernel 1: h = text_emb @ W_in + b_in + freq_pe   (fp32 WMMA 16x16x4)
//
// Block = 256 threads (8 waves). Block computes a 128x16 output tile:
//   - wave 0 DMAs the 512x16 fp32 W_in column panel (32 KB) into LDS with the
//     Tensor Data Mover (tensor_load_to_lds + s_wait_tensorcnt)
//   - each wave then computes one 16x16 tile, A-fragments from global
//     (text_emb streams once, L2-resident), B-fragments from LDS.
// ---------------------------------------------------------------------------
__global__ __launch_bounds__(256) void gemm_pe_kernel(
    const float* __restrict__ A,     // text_emb  (8192, 512)
    const float* __restrict__ W,     // W_in      (512, 512)
    const float* __restrict__ bias,  // b_in      (512)
    const float* __restrict__ freq,  // freq_pe   (2048, 512)
    float* __restrict__ H)           // out h     (8192, 512)
{
    __shared__ float bpanel[E_DIM * 16];          // 32 KB: W[:, n0..n0+15]

    const int lane  = threadIdx.x & 31;
    const int wib   = threadIdx.x >> 5;
    const int tileN = blockIdx.x;                 // 0..31
    const int tileM = blockIdx.y * 8 + wib;       // 0..511
    const int m     = lane & 15;
    const int half  = lane >> 4;                  // 0: K 0..1, 1: K 2..3

    // --- TDM: DMA W_in[:, tileN*16 .. +16) into LDS (issued by wave 0) -----
    if (wib == 0) {
        const unsigned long long gaddr =
            (unsigned long long)(const void*)(W + (size_t)tileN * 16);
        const unsigned lds_addr = (unsigned)(size_t)(void*)&bpanel[0];

        v4u g0;
        g0.x = 1u;                                        // count=1, user mode
        g0.y = lds_addr;                                  // LDS byte address
        g0.z = (unsigned)(gaddr & 0xFFFFFFFFu);           // global_addr[31:0]
        g0.w = (unsigned)((gaddr >> 32) & 0x1FFFFFFu)     // global_addr[56:32]
             | 0x80000000u;                               // type=2 ("image")

        v8i g1;
        g1[0] = 0x00020000;          // workgroup_mask=0, data_size=2 (4B)
        g1[1] = (E_DIM & 0xFFFF) << 16;                   // tensor_dim0=512
        g1[2] = (E_DIM & 0xFFFF) << 16;                   // tensor_dim1=512
        g1[3] = 16 << 16;                                 // tile_dim0=16
        g1[4] = E_DIM;               // tile_dim1=512, tile_dim2=0
        g1[5] = D_DIM;               // tensor_dim0_stride=512 (elements)
        g1[6] = 0;                   // stride hi / dim1_stride lo
        g1[7] = 0;                   // dim1_stride hi

        v4i g2 = {0, 0, 0, 0};       // 2-D tensor: groups 2/3 unused
        v4i g3 = {0, 0, 0, 0};
        v8i g4 = {0, 0, 0, 0, 0, 0, 0, 0};   // extra group (clang-23 6-arg form)

        __builtin_amdgcn_tensor_load_to_lds(g0, g1, g2, g3, g4, 0);
        __builtin_amdgcn_s_wait_tensorcnt(0);
    }
    __syncthreads();

    // --- WMMA main loop -----------------------------------------------------
    const float* Arow = A + (size_t)(tileM * 16 + m) * E_DIM;

    v8f acc = {0.f, 0.f, 0.f, 0.f, 0.f, 0.f, 0.f, 0.f};

#pragma unroll 4
    for (int k0 = 0; k0 < E_DIM; k0 += 4) {
        const int kk = k0 + half * 2;
        v2f a, b;
        a.x = Arow[kk + 0];
        a.y = Arow[kk + 1];
        b.x = bpanel[(kk + 0) * 16 + m];
        b.y = bpanel[(kk + 1) * 16 + m];
        acc = __builtin_amdgcn_wmma_f32_16x16x4_f32(
            /*neg_a=*/false, a, /*neg_b=*/false, b,
            /*c_mod=*/(short)0, acc, /*reuse_a=*/false, /*reuse_b=*/false);
    }

    const int c  = tileN * 16 + m;
    const float bc = bias[c];
#pragma unroll
    for (int v = 0; v < 8; ++v) {
        const int r = tileM * 16 + v + half * 8;
        const int t = r & (T_LEN - 1);        // row -> time index (b*T + t)
        H[(size_t)r * D_DIM + c] = acc[v] + bc + freq[(size_t)t * D_DIM + c];
    }
}

// ---------------------------------------------------------------------------
// Kernel 2: one S4 layer, in-place update of H.
// Thread layout: 16 lanes (n=0..15) per (b,d) group, 2 groups per wave.
// Each step: x = Abar*x + Bbar*u; y = sum_n(x*C) via shfl_xor butterfly;
// h <- h + gelu(y + Dsk*h).
// ---------------------------------------------------------------------------
__global__ __launch_bounds__(256) void s4_layer_kernel(
    float* __restrict__ H,               // (B, T, D) in-place
    const float* __restrict__ A_log,     // (D, N) for this layer
    const float* __restrict__ B_ssm,     // (D, N)
    const float* __restrict__ C_ssm,     // (D, N)
    const float* __restrict__ log_dt,    // (D)
    const float* __restrict__ D_skip)    // (D)
{
    const int tid = blockIdx.x * blockDim.x + threadIdx.x;
    const int n   = tid & 15;
    const int g   = tid >> 4;             // (b,d) group index
    const int d   = g & (D_DIM - 1);
    const int b   = g >> 9;               // g / 512

    const float dt   = expf(log_dt[d]);
    const float a    = -expf(A_log[d * N_ST + n]);
    const float abar = expf(dt * a);
    const float bbar = (abar - 1.0f) / a * B_ssm[d * N_ST + n];
    const float cn   = C_ssm[d * N_ST + n];
    const float dsk  = D_skip[d];

    float x = 0.0f;
    float* hp = H + (size_t)b * T_LEN * D_DIM + d;

    for (int t = 0; t < T_LEN; ++t) {
        const float u = hp[(size_t)t * D_DIM];
        x = fmaf(abar, x, bbar * u);
        float p = x * cn;
        // reduce over the 16 n-lanes of this group (stays within half-wave)
        p += __shfl_xor(p, 1, 32);
        p += __shfl_xor(p, 2, 32);
        p += __shfl_xor(p, 4, 32);
        p += __shfl_xor(p, 8, 32);
        const float y = p + dsk * u;
        // tanh-approx gelu (jax.nn.gelu default)
        const float g3 = 0.7978845608028654f * (y + 0.044715f * y * y * y);
        const float gl = 0.5f * y * (1.0f + tanhf(g3));
        if (n == 0) hp[(size_t)t * D_DIM] = u + gl;
    }
}

// ---------------------------------------------------------------------------
// Kernel 3: heads. One block per (b,t) row: shared mean/var, two LN+dot heads.
// out[0 .. BT)    = f0
// out[BT .. 2BT)  = en
// ---------------------------------------------------------------------------
__global__ __launch_bounds__(256) void head_kernel(
    const float* __restrict__ H,
    const float* __restrict__ s_f0, const float* __restrict__ bln_f0,
    const float* __restrict__ W_f0, const float* __restrict__ b_f0,
    const float* __restrict__ s_en, const float* __restrict__ bln_en,
    const float* __restrict__ W_en, const float* __restrict__ b_en,
    float* __restrict__ out)
{
    __shared__ float red0[256];
    __shared__ float red1[256];

    const int r   = blockIdx.x;           // row in [0, B*T)
    const int tid = threadIdx.x;
    const float* hr = H + (size_t)r * D_DIM;

    const float h0 = hr[tid];
    const float h1 = hr[tid + 256];

    red0[tid] = h0 + h1;
    red1[tid] = h0 * h0 + h1 * h1;
    __syncthreads();
    for (int off = 128; off > 0; off >>= 1) {
        if (tid < off) {
            red0[tid] += red0[tid + off];
            red1[tid] += red1[tid + off];
        }
        __syncthreads();
    }
    const float mu   = red0[0] * (1.0f / (float)D_DIM);
    const float var  = red1[0] * (1.0f / (float)D_DIM) - mu * mu;
    const float rstd = rsqrtf(var + 1e-5f);
    __syncthreads();

    const float n0 = (h0 - mu) * rstd;
    const float n1 = (h1 - mu) * rstd;
    const float f0p = (n0 * s_f0[tid]       + bln_f0[tid])       * W_f0[tid] +
                      (n1 * s_f0[tid + 256] + bln_f0[tid + 256]) * W_f0[tid + 256];
    const float enp = (n0 * s_en[tid]       + bln_en[tid])       * W_en[tid] +
                      (n1 * s_en[tid + 256] + bln_en[tid + 256]) * W_en[tid + 256];

    red0[tid] = f0p;
    red1[tid] = enp;
    __syncthreads();
    for (int off = 128; off > 0; off >>= 1) {
        if (tid < off) {
            red0[tid] += red0[tid + off];
            red1[tid] += red1[tid + off];
        }
        __syncthreads();
    }
    if (tid == 0) {
        out[r]          = red0[0] + b_f0[0];
        out[M_ROWS + r] = red1[0] + b_en[0];
    }
}

// ---------------------------------------------------------------------------
extern "C" void kernel_launch(void* const* d_in, const int* in_sizes, int n_in,
                              void* d_out, int out_size, void* d_ws, size_t ws_size,
                              hipStream_t stream)
{
    const float* text_emb = (const float*)d_in[0];   // (B,T,E)
    const float* W_in     = (const float*)d_in[1];   // (E,D)
    const float* b_in     = (const float*)d_in[2];   // (D)
    const float* freq_pe  = (const float*)d_in[3];   // (NF,D)
    const float* A_log    = (const float*)d_in[4];   // (L,D,N)
    const float* B_ssm    = (const float*)d_in[5];   // (L,D,N)
    const float* C_ssm    = (const float*)d_in[6];   // (L,D,N)
    const float* log_dt   = (const float*)d_in[7];   // (L,D)
    const float* D_skip   = (const float*)d_in[8];   // (L,D)
    const float* ln_f0_s  = (const float*)d_in[9];
    const float* ln_f0_b  = (const float*)d_in[10];
    const float* W_f0     = (const float*)d_in[11];
    const float* b_f0     = (const float*)d_in[12];
    const float* ln_en_s  = (const float*)d_in[13];
    const float* ln_en_b  = (const float*)d_in[14];
    const float* W_en     = (const float*)d_in[15];
    const float* b_en     = (const float*)d_in[16];

    float* H   = (float*)d_ws;                       // (B,T,D) = 16 MiB
    float* out = (float*)d_out;                      // f0 (B*T) ++ en (B*T)

    // 1) GEMM + bias + positional add.
    //    grid.x = N tiles (32), grid.y = M/128 (64); block = 8 waves.
    {
        dim3 grid(D_DIM / 16, M_ROWS / 128);
        gemm_pe_kernel<<<grid, 256, 0, stream>>>(text_emb, W_in, b_in, freq_pe, H);
    }

    // 2) Four S4 scan layers, in place on H
    {
        const int threads = B_N * D_DIM * N_ST;      // 32768
        const int blocks  = threads / 256;           // 128
        for (int l = 0; l < L_NUM; ++l) {
            s4_layer_kernel<<<blocks, 256, 0, stream>>>(
                H,
                A_log  + (size_t)l * D_DIM * N_ST,
                B_ssm  + (size_t)l * D_DIM * N_ST,
                C_ssm  + (size_t)l * D_DIM * N_ST,
                log_dt + (size_t)l * D_DIM,
                D_skip + (size_t)l * D_DIM);
        }
    }

    // 3) LayerNorm + linear heads
    head_kernel<<<M_ROWS, 256, 0, stream>>>(
        H, ln_f0_s, ln_f0_b, W_f0, b_f0, ln_en_s, ln_en_b, W_en, b_en, out);
}